// BreakthroughSNN_34505767256902
// MI455X (gfx1250) — compile-verified
//
#include <hip/hip_runtime.h>

typedef __attribute__((ext_vector_type(2))) float v2f;
typedef __attribute__((ext_vector_type(8))) float v8f;

#define NWG      64
#define THREADS  128
#define S_LEN    128
#define DM       1024
#define B_SZ     32
#define T_NUM    4
#define VOCAB    32000

#define DECAYF   0.60653065971263342f   /* exp(-1/2) */

/* d_ws float layout */
#define WS_STATES 256                         /* [0]=cnt [1]=gen [8..72)=err partials */
#define STATES_SZ (S_LEN * 2 * B_SZ * DM)     /* 8388608 */
#define WS_ERR0  (WS_STATES + STATES_SZ)
#define WS_ERR1  (WS_ERR0 + B_SZ * DM)
#define WS_TD    (WS_ERR1 + B_SZ * DM)
#define WS_FACT  (WS_TD   + B_SZ * DM)

/* ---- f32 WMMA 16x16x4: D = A(16x4) * B(4x16) + C ----
   A: lane L(0..15)->M=L K={k,k+1}; lane L+16 -> M=L K={k+2,k+3}
   B: lane L(0..15)->N=L K={k,k+1}; lane L+16 -> N=L K={k+2,k+3}
   C: vgpr r, lanes 0-15 -> M=r, lanes 16-31 -> M=r+8; N=lane&15
   Ap/Bp must already include per-lane koff = 2*(lane>>4) AND the wave's k0,
   so the trip count KLEN is a compile-time constant (uniform loop, no
   exec-masked loop idiom around the WMMA chain).                          */
template <int KLEN>
__device__ inline v8f mm_tile(const float* __restrict__ Ap,
                              const float* __restrict__ Bp) {
  v8f acc = {};
#pragma unroll 8
  for (int k = 0; k < KLEN; k += 4) {
    v2f a = *(const v2f*)(Ap + k);
    v2f b = *(const v2f*)(Bp + k);
    acc = __builtin_amdgcn_wmma_f32_16x16x4_f32(false, a, false, b,
                                                (short)0, acc, false, false);
  }
  return acc;
}

/* LIF (eval mode): mem = mem*decay + x + bias; spike = mem>=1; hard reset */
__device__ inline void lif_step(v8f& mem, const v8f& x, float bias, v8f& spk) {
#pragma unroll
  for (int r = 0; r < 8; ++r) {
    float mm = mem[r] * DECAYF + x[r] + bias;
    float sp = (mm >= 1.0f) ? 1.0f : 0.0f;
    mem[r] = mm * (1.0f - sp);
    spk[r] = sp;
  }
}
/* LIF with zero matmul input (down-pass inf layers: err == 0 exactly) */
__device__ inline void lif_decay(v8f& mem, float bias) {
#pragma unroll
  for (int r = 0; r < 8; ++r) {
    float mm = mem[r] * DECAYF + bias;
    float sp = (mm >= 1.0f) ? 1.0f : 0.0f;
    mem[r] = mm * (1.0f - sp);
  }
}

/* combine the two K-halves (kh=1 wave -> LDS -> kh=0 wave) */
__device__ inline void reduce3(float* red, int m, int kh, int lane,
                               v8f& a0, v8f& a1, v8f& a2) {
  __syncthreads();
  if (kh == 1) {
#pragma unroll
    for (int r = 0; r < 8; ++r) {
      red[(m * 3 + 0) * 256 + r * 32 + lane] = a0[r];
      red[(m * 3 + 1) * 256 + r * 32 + lane] = a1[r];
      red[(m * 3 + 2) * 256 + r * 32 + lane] = a2[r];
    }
  }
  __syncthreads();
  if (kh == 0) {
#pragma unroll
    for (int r = 0; r < 8; ++r) {
      a0[r] += red[(m * 3 + 0) * 256 + r * 32 + lane];
      a1[r] += red[(m * 3 + 1) * 256 + r * 32 + lane];
      a2[r] += red[(m * 3 + 2) * 256 + r * 32 + lane];
    }
  }
}
__device__ inline void reduce1(float* red, int m, int kh, int lane, v8f& a0) {
  __syncthreads();
  if (kh == 1) {
#pragma unroll
    for (int r = 0; r < 8; ++r) red[m * 3 * 256 + r * 32 + lane] = a0[r];
  }
  __syncthreads();
  if (kh == 0) {
#pragma unroll
    for (int r = 0; r < 8; ++r) a0[r] += red[m * 3 * 256 + r * 32 + lane];
  }
}

/* device-wide barrier across the persistent grid (counter + generation) */
__device__ inline void grid_barrier(unsigned* bar) {
  __syncthreads();
  if (threadIdx.x == 0) {
    __threadfence();
    unsigned* cnt = bar;
    unsigned* gen = bar + 1;
    unsigned g = __hip_atomic_load(gen, __ATOMIC_RELAXED, __HIP_MEMORY_SCOPE_AGENT);
    unsigned prev = __hip_atomic_fetch_add(cnt, 1u, __ATOMIC_ACQ_REL, __HIP_MEMORY_SCOPE_AGENT);
    if (prev == NWG - 1u) {
      __hip_atomic_store(cnt, 0u, __ATOMIC_RELAXED, __HIP_MEMORY_SCOPE_AGENT);
      __hip_atomic_fetch_add(gen, 1u, __ATOMIC_RELEASE, __HIP_MEMORY_SCOPE_AGENT);
    } else {
      while (__hip_atomic_load(gen, __ATOMIC_ACQUIRE, __HIP_MEMORY_SCOPE_AGENT) == g)
        __builtin_amdgcn_s_sleep(2);
    }
    __threadfence();
  }
  __syncthreads();
}

__global__ void snn_zero(float* __restrict__ ws, unsigned n) {
  unsigned i = blockIdx.x * blockDim.x + threadIdx.x;
  if (i < n) ws[i] = 0.0f;
}

__global__ void __launch_bounds__(THREADS) snn_recurrent(
    const int* __restrict__ ids, const float* __restrict__ emb,
    const float* __restrict__ enc_w, const float* __restrict__ enc_b,
    const float* __restrict__ gen_w, const float* __restrict__ gen_b,
    const float* __restrict__ inf_w, const float* __restrict__ inf_b,
    float* __restrict__ ws) {
  __shared__ float red[2 * 3 * 256];
  __shared__ float esum[THREADS];

  unsigned* bar  = (unsigned*)ws;
  float* states  = ws + WS_STATES;
  float* err0b   = ws + WS_ERR0;
  float* err1b   = ws + WS_ERR1;
  float* tdb     = ws + WS_TD;
  float* factb   = ws + WS_FACT;

  const int wg   = blockIdx.x;
  const int tid  = threadIdx.x;
  const int lane = tid & 31;
  const int wv   = tid >> 5;
  const int m    = wv & 1;        /* M tile: batch rows [m*16, m*16+16) */
  const int kh   = wv >> 1;       /* K half */
  const int hi   = lane >> 4;
  const int nl   = lane & 15;
  const int col  = wg * 16 + nl;  /* output column this WG owns        */
  const int rowA = m * 16 + nl;   /* batch row for A-matrix loads       */
  const int rbase = m * 16 + hi * 8; /* C/D tile row base               */
  /* per-lane K start: wave K-half + A/B lane sub-offset */
  const int kbase = kh * (DM / 2) + 2 * hi;

  const float* Benc = enc_w + (size_t)col * DM + kbase;
  const float* Bg0  = gen_w + (size_t)col * DM + kbase;
  const float* Bg1  = gen_w + (size_t)(DM * DM) + (size_t)col * DM + kbase;
  const float* Bi0  = inf_w + (size_t)col * DM + kbase;
  const float* Bi1  = inf_w + (size_t)(DM * DM) + (size_t)col * DM + kbase;

  const float b_enc = enc_b[col];
  const float b_g0  = gen_b[col];
  const float b_g1  = gen_b[DM + col];
  const float b_i0  = inf_b[col];
  const float b_i1  = inf_b[DM + col];

  v8f mem_enc = {}, mem_g0 = {}, mem_g1 = {}, mem_i0 = {}, mem_i1 = {};
  v8f pred1 = {}, final_acc = {};
  float err_sum = 0.0f;

  for (int t = 0; t < T_NUM; ++t) {
    for (int s = 0; s < S_LEN; ++s) {
      /* phase A: enc, gen-up0, gen-up1 (all independent of this step) */
      const int id = ids[rowA * S_LEN + s];
      v8f aE  = mm_tile<DM / 2>(emb + (size_t)id * DM + kbase, Benc);
      v8f aG0 = mm_tile<DM / 2>(states + (size_t)((s * 2 + 0) * B_SZ + rowA) * DM + kbase, Bg0);
      v8f aG1 = mm_tile<DM / 2>(states + (size_t)((s * 2 + 1) * B_SZ + rowA) * DM + kbase, Bg1);
      reduce3(red, m, kh, lane, aE, aG0, aG1);
      if (kh == 0) {
        v8f bu, p0;
        lif_step(mem_enc, aE,  b_enc, bu);
        lif_step(mem_g0,  aG0, b_g0,  p0);
        lif_step(mem_g1,  aG1, b_g1,  pred1);
#pragma unroll
        for (int r = 0; r < 8; ++r) {
          float e = bu[r] - p0[r]; e = e > 0.0f ? e : 0.0f;
          err_sum += e;
          err0b[(rbase + r) * DM + col] = e;
        }
      }
      grid_barrier(bar);
      /* inf0: su0 = lif(err0 @ inf_w0^T); ns0 = state0 + su0; err1 = relu(ns0 - pred1) */
      v8f aI0 = mm_tile<DM / 2>(err0b + (size_t)rowA * DM + kbase, Bi0);
      reduce1(red, m, kh, lane, aI0);
      if (kh == 0) {
        v8f su0;
        lif_step(mem_i0, aI0, b_i0, su0);
        float* st0 = states + (size_t)(s * 2 + 0) * B_SZ * DM;
#pragma unroll
        for (int r = 0; r < 8; ++r) {
          float ns = st0[(rbase + r) * DM + col] + su0[r];
          st0[(rbase + r) * DM + col] = ns;
          float e = ns - pred1[r]; e = e > 0.0f ? e : 0.0f;
          err_sum += e;
          err1b[(rbase + r) * DM + col] = e;
        }
      }
      grid_barrier(bar);
      /* inf1: su1; ns1 = state1 + su1 (in place -> td for down pass) */
      v8f aI1 = mm_tile<DM / 2>(err1b + (size_t)rowA * DM + kbase, Bi1);
      reduce1(red, m, kh, lane, aI1);
      if (kh == 0) {
        v8f su1;
        lif_step(mem_i1, aI1, b_i1, su1);
        float* st1 = states + (size_t)(s * 2 + 1) * B_SZ * DM;
#pragma unroll
        for (int r = 0; r < 8; ++r) st1[(rbase + r) * DM + col] += su1[r];
      }
      grid_barrier(bar);
      /* down gen1: pred1d = lif(ns1 @ gen_w1^T); inf1 decays on zero input */
      v8f aD1 = mm_tile<DM / 2>(states + (size_t)((s * 2 + 1) * B_SZ + rowA) * DM + kbase, Bg1);
      reduce1(red, m, kh, lane, aD1);
      if (kh == 0) {
        v8f p1d;
        lif_step(mem_g1, aD1, b_g1, p1d);
        lif_decay(mem_i1, b_i1);
#pragma unroll
        for (int r = 0; r < 8; ++r) tdb[(rbase + r) * DM + col] = p1d[r];
      }
      grid_barrier(bar);
      /* down gen0: pred = lif(td @ gen_w0^T); collect at s == S-1 */
      v8f aD0 = mm_tile<DM / 2>(tdb + (size_t)rowA * DM + kbase, Bg0);
      reduce1(red, m, kh, lane, aD0);
      if (kh == 0) {
        v8f p0d;
        lif_step(mem_g0, aD0, b_g0, p0d);
        lif_decay(mem_i0, b_i0);
        if (s == S_LEN - 1) {
#pragma unroll
          for (int r = 0; r < 8; ++r) final_acc[r] += p0d[r];
        }
      }
      /* no barrier needed here: 3 barriers separate the next writers of tdb */
    }
  }
  /* epilogue: final activation (mean over T) + deterministic err reduction */
  if (kh == 0) {
#pragma unroll
    for (int r = 0; r < 8; ++r)
      factb[(rbase + r) * DM + col] = final_acc[r] * 0.25f;
  }
  esum[tid] = err_sum;
  __syncthreads();
  if (tid == 0) {
    float ssum = 0.0f;
    for (int i = 0; i < THREADS; ++i) ssum += esum[i];
    ws[8 + wg] = ssum;
  }
}

__global__ void __launch_bounds__(128) snn_logits(
    const float* __restrict__ ws,
    const float* __restrict__ out_w, const float* __restrict__ out_b,
    float* __restrict__ out) {
  const float* factb = ws + WS_FACT;
  const int tid  = threadIdx.x;
  const int lane = tid & 31;
  const int wv   = tid >> 5;
  const int m    = wv & 1;
  const int nsub = wv >> 1;
  const int hi   = lane >> 4;
  const int nl   = lane & 15;
  const int koff = 2 * hi;
  const int rowA = m * 16 + nl;
  const int col  = blockIdx.x * 32 + nsub * 16 + nl;

  v8f acc = mm_tile<DM>(factb + (size_t)rowA * DM + koff,
                        out_w + (size_t)col * DM + koff);
  const float bias = out_b[col];
  const int rbase = m * 16 + hi * 8;
#pragma unroll
  for (int r = 0; r < 8; ++r)
    out[(size_t)(rbase + r) * VOCAB + col] = acc[r] + bias;

  if (blockIdx.x == 0 && tid == 0) {
    float ssum = 0.0f;
    for (int i = 0; i < NWG; ++i) ssum += ws[8 + i];
    out[(size_t)B_SZ * VOCAB] = ssum * (1.0f / (T_NUM * S_LEN * B_SZ));
  }
}

extern "C" void kernel_launch(void* const* d_in, const int* in_sizes, int n_in,
                              void* d_out, int out_size, void* d_ws, size_t ws_size,
                              hipStream_t stream) {
  const int*   ids   = (const int*)  d_in[0];
  const float* emb   = (const float*)d_in[1];
  const float* enc_w = (const float*)d_in[2];
  const float* enc_b = (const float*)d_in[3];
  const float* gen_w = (const float*)d_in[4];
  const float* gen_b = (const float*)d_in[5];
  const float* inf_w = (const float*)d_in[6];
  const float* inf_b = (const float*)d_in[7];
  const float* out_w = (const float*)d_in[8];
  const float* out_b = (const float*)d_in[9];
  float* ws  = (float*)d_ws;
  float* out = (float*)d_out;

  const unsigned nz = WS_STATES + STATES_SZ;  /* counters + states must be zero */
  snn_zero<<<(nz + 255) / 256, 256, 0, stream>>>(ws, nz);
  snn_recurrent<<<NWG, THREADS, 0, stream>>>(ids, emb, enc_w, enc_b,
                                             gen_w, gen_b, inf_w, inf_b, ws);
  snn_logits<<<VOCAB / 32, 128, 0, stream>>>(ws, out_w, out_b, out);
}